// TensorCSPNet_22643067584704
// MI455X (gfx1250) — compile-verified
//
#include <hip/hip_runtime.h>
#include <hip/hip_bf16.h>

typedef __attribute__((ext_vector_type(2))) float v2f;
typedef __attribute__((ext_vector_type(8))) float v8f;

// Problem constants
#define NB   256
#define NF   9
#define NC   22
#define NT   1000
#define NP   4
#define ND   36          // F*P
#define ND1  20
#define ND2  18
#define NTC  16
#define NOUT 4
#define TP   250         // T/P
#define EPSV 1e-4f
#define NMAT (NB*ND)     // 9216
#define LDJ  25          // LDS leading dim (odd -> conflict-free-ish)
#define NSWEEP 10

// intra-wave LDS phase fence: wave32 DS ops are in-order; this only stops
// compiler code motion (zero runtime cost, unlike s_barrier).
#define WFENCE() __builtin_amdgcn_wave_barrier()

// ---------------------------------------------------------------------------
// workspace layout (floats)
#define WS_COV 0                         // 9216 * 484
#define WS_S2  (NMAT*484)                // 9216 * 324
#define WS_XL  (WS_S2 + NMAT*324)        // 256 * 11664
#define WS_GL  (WS_XL + NB*11664)        // 324
#define WS_GM  (WS_GL + 324)             // 324

// ---------------------------------------------------------------------------
__global__ void zero_gl_kernel(float* __restrict__ gl) {
    int t = threadIdx.x;
    if (t < 324) gl[t] = 0.0f;
}

// ---------------------------------------------------------------------------
// Covariance via V_WMMA_F32_16X16X4_F32.  One wave per (b,f,p); 2 waves/block.
// Centered 22x250 data staged in LDS padded to 32 rows x 252 cols (zeros), so
// the inner loop is branch-free: 2x ds_load_b64 + 4x v_wmma per K-step of 4.
#define TPW  252
#define ROWS 32
__global__ void cov_kernel(const float* __restrict__ x, float* __restrict__ covout) {
    __shared__ float Xs[2][ROWS * TPW];         // 64512 B
    const int wv = threadIdx.x >> 5, lane = threadIdx.x & 31;
    const int idx = blockIdx.x * 2 + wv;        // = b*36 + d, d = f*4+p
    const int b = idx / ND, d = idx - b * ND;
    const int f = d >> 2, p = d & 3;
    const float* src = x + ((size_t)(b * NF + f) * NC) * NT + p * TP;
    float* X = Xs[wv];

    // zero pad rows 22..31 (M-pad for the 2nd fragment)
    for (int i = lane; i < (ROWS - NC) * TPW; i += 32) X[NC * TPW + i] = 0.f;
    // stage 22x250
    for (int i = lane; i < NC * TP; i += 32) {
        int c = i / TP, t = i - c * TP;
        X[c * TPW + t] = src[(size_t)c * NT + t];
    }
    WFENCE();
    // center each row; zero K-pad columns 250,251
    if (lane < NC) {
        float s = 0.f;
        for (int t = 0; t < TP; ++t) s += X[lane * TPW + t];
        float mu = s * (1.0f / TP);
        for (int t = 0; t < TP; ++t) X[lane * TPW + t] -= mu;
        X[lane * TPW + 250] = 0.f;
        X[lane * TPW + 251] = 0.f;
    }
    WFENCE();

    v8f c00 = {}, c01 = {}, c10 = {}, c11 = {};
    const int r0 = lane & 15;            // rows 0..15
    const int r1 = 16 + (lane & 15);     // rows 16..31 (zero padded)
    const int ko = 2 * (lane >> 4);      // K sub-offset per fp32 A/B layout
    for (int kk = 0; kk < 63; ++kk) {
        int k = kk * 4 + ko;
        v2f a0 = *(const v2f*)&X[r0 * TPW + k];
        v2f a1 = *(const v2f*)&X[r1 * TPW + k];
        // identical lane layout for A(16x4) and B(4x16)=A^T here -> Gram tiles
        c00 = __builtin_amdgcn_wmma_f32_16x16x4_f32(false, a0, false, a0, (short)0, c00, false, false);
        c01 = __builtin_amdgcn_wmma_f32_16x16x4_f32(false, a0, false, a1, (short)0, c01, false, false);
        c10 = __builtin_amdgcn_wmma_f32_16x16x4_f32(false, a1, false, a0, (short)0, c10, false, false);
        c11 = __builtin_amdgcn_wmma_f32_16x16x4_f32(false, a1, false, a1, (short)0, c11, false, false);
    }
    float* out = covout + (size_t)idx * (NC * NC);
    const float inv = 1.0f / (TP - 1);
    const int nn = lane & 15, mhi = 8 * (lane >> 4);
    #pragma unroll
    for (int r = 0; r < 8; ++r) {
        int m0 = mhi + r;
        int m1 = 16 + mhi + r;
        int n0 = nn, n1 = 16 + nn;
        if (m0 < NC && n0 < NC) out[m0 * NC + n0] = c00[r] * inv;
        if (m0 < NC && n1 < NC) out[m0 * NC + n1] = c01[r] * inv;
        if (m1 < NC && n0 < NC) out[m1 * NC + n0] = c10[r] * inv;
        if (m1 < NC && n1 < NC) out[m1 * NC + n1] = c11[r] * inv;
    }
}

// ---------------------------------------------------------------------------
// Wave-parallel cyclic Jacobi eigh for n<=20 symmetric matrix in LDS (ld=LDJ).
// Each wave owns its matrices: only compiler fences needed (DS is in-order
// within a wave on CDNA5).  On exit: diag(A)=eigenvalues, V cols=eigenvectors.
__device__ __forceinline__ void jacobi_sym(float* A, float* V, int n, int lane) {
    if (lane < n)
        for (int i = 0; i < n; ++i) V[i * LDJ + lane] = (i == lane) ? 1.f : 0.f;
    WFENCE();
    for (int sweep = 0; sweep < NSWEEP; ++sweep) {
        for (int p = 0; p < n - 1; ++p) {
            for (int q = p + 1; q < n; ++q) {
                float apq = A[p * LDJ + q];
                float app = A[p * LDJ + p];
                float aqq = A[q * LDJ + q];
                float c = 1.f, s = 0.f;
                if (fabsf(apq) > 1e-12f) {
                    float tau = (aqq - app) / (2.f * apq);
                    float t = (tau >= 0.f ? 1.f : -1.f) / (fabsf(tau) + sqrtf(1.f + tau * tau));
                    c = 1.f / sqrtf(1.f + t * t);
                    s = t * c;
                }
                WFENCE();
                if (lane < n) {               // column update: A <- A*J, V <- V*J
                    int j = lane;
                    float ajp = A[j * LDJ + p], ajq = A[j * LDJ + q];
                    A[j * LDJ + p] = c * ajp - s * ajq;
                    A[j * LDJ + q] = s * ajp + c * ajq;
                    float vjp = V[j * LDJ + p], vjq = V[j * LDJ + q];
                    V[j * LDJ + p] = c * vjp - s * vjq;
                    V[j * LDJ + q] = s * vjp + c * vjq;
                }
                WFENCE();
                if (lane < n) {               // row update: A <- J^T * A
                    int j = lane;
                    float apj = A[p * LDJ + j], aqj = A[q * LDJ + j];
                    A[p * LDJ + j] = c * apj - s * aqj;
                    A[q * LDJ + j] = s * apj + c * aqj;
                }
                WFENCE();
            }
        }
    }
}

// ---------------------------------------------------------------------------
// Per (b,d): S1 = W1 cov W1^T (20x20); ReEig; S2 = W2 S1' W2^T (18x18);
// store S2; LogEig(S2) accumulated into Gl (mean of logs).
__global__ void bimap_kernel(const float* __restrict__ covg,
                             const float* __restrict__ W1g,
                             const float* __restrict__ W2g,
                             float* __restrict__ S2g,
                             float* __restrict__ Gl) {
    __shared__ float MaS[4][LDJ * LDJ], MbS[4][LDJ * LDJ],
                     MvS[4][LDJ * LDJ], MwS[4][LDJ * LDJ];   // 40 KB
    const int wv = threadIdx.x >> 5, lane = threadIdx.x & 31;
    const int idx = blockIdx.x * 4 + wv;
    const int d = idx % ND;
    float *Ma = MaS[wv], *Mb = MbS[wv], *Mv = MvS[wv], *Mw = MwS[wv];

    for (int i = lane; i < NC * NC; i += 32)
        Ma[(i / NC) * LDJ + (i % NC)] = covg[(size_t)idx * (NC * NC) + i];
    for (int i = lane; i < ND1 * NC; i += 32)
        Mw[(i / NC) * LDJ + (i % NC)] = W1g[(size_t)d * (ND1 * NC) + i];
    WFENCE();
    // T = cov * W1^T  (22x20) -> Mb
    if (lane < ND1)
        for (int i = 0; i < NC; ++i) {
            float s = 0.f;
            for (int t = 0; t < NC; ++t) s += Ma[i * LDJ + t] * Mw[lane * LDJ + t];
            Mb[i * LDJ + lane] = s;
        }
    WFENCE();
    // S1 = W1 * T  (20x20) -> Ma
    if (lane < ND1)
        for (int i = 0; i < ND1; ++i) {
            float s = 0.f;
            for (int t = 0; t < NC; ++t) s += Mw[i * LDJ + t] * Mb[t * LDJ + lane];
            Ma[i * LDJ + lane] = s;
        }
    WFENCE();
    jacobi_sym(Ma, Mv, ND1, lane);
    // ReEig: Mb = V * max(w,EPS) * V^T
    if (lane < ND1)
        for (int i = 0; i < ND1; ++i) {
            float s = 0.f;
            for (int t = 0; t < ND1; ++t)
                s += Mv[i * LDJ + t] * fmaxf(Ma[t * LDJ + t], EPSV) * Mv[lane * LDJ + t];
            Mb[i * LDJ + lane] = s;
        }
    WFENCE();
    for (int i = lane; i < ND2 * ND1; i += 32)
        Mw[(i / ND1) * LDJ + (i % ND1)] = W2g[(size_t)d * (ND2 * ND1) + i];
    WFENCE();
    // T = S1' * W2^T (20x18) -> Ma
    if (lane < ND2)
        for (int i = 0; i < ND1; ++i) {
            float s = 0.f;
            for (int t = 0; t < ND1; ++t) s += Mb[i * LDJ + t] * Mw[lane * LDJ + t];
            Ma[i * LDJ + lane] = s;
        }
    WFENCE();
    // S2 = W2 * T (18x18) -> Mb
    if (lane < ND2)
        for (int i = 0; i < ND2; ++i) {
            float s = 0.f;
            for (int t = 0; t < ND1; ++t) s += Mw[i * LDJ + t] * Ma[t * LDJ + lane];
            Mb[i * LDJ + lane] = s;
        }
    WFENCE();
    // store S2
    if (lane < ND2)
        for (int i = 0; i < ND2; ++i)
            S2g[(size_t)idx * (ND2 * ND2) + i * ND2 + lane] = Mb[i * LDJ + lane];
    WFENCE();
    jacobi_sym(Mb, Mv, ND2, lane);
    // L = V * log(max(w,1e-10)) * V^T ; Gl += L / NMAT
    if (lane < ND2)
        for (int i = 0; i < ND2; ++i) {
            float s = 0.f;
            for (int t = 0; t < ND2; ++t)
                s += Mv[i * LDJ + t] * logf(fmaxf(Mb[t * LDJ + t], 1e-10f)) * Mv[lane * LDJ + t];
            atomicAdd(&Gl[i * ND2 + lane], s * (1.0f / NMAT));
        }
}

// ---------------------------------------------------------------------------
// Gm = V * exp(-0.5 * w) * V^T  from eigh of the batch log-mean Gl
__global__ void gm_kernel(const float* __restrict__ Gl, float* __restrict__ Gm) {
    __shared__ float Ma[LDJ * LDJ], Mv[LDJ * LDJ];
    const int lane = threadIdx.x;
    for (int i = lane; i < ND2 * ND2; i += 32)
        Ma[(i / ND2) * LDJ + (i % ND2)] = Gl[i];
    WFENCE();
    jacobi_sym(Ma, Mv, ND2, lane);
    if (lane < ND2)
        for (int i = 0; i < ND2; ++i) {
            float s = 0.f;
            for (int t = 0; t < ND2; ++t)
                s += Mv[i * LDJ + t] * expf(-0.5f * Ma[t * LDJ + t]) * Mv[lane * LDJ + t];
            Gm[i * ND2 + lane] = s;
        }
}

// ---------------------------------------------------------------------------
// Sf = Gm S2 Gm; fused ReEig+LogEig (same eigenvectors): Xl = V log(max(w,EPS)) V^T
__global__ void bn_kernel(const float* __restrict__ S2g,
                          const float* __restrict__ Gm,
                          float* __restrict__ Xlg) {
    __shared__ float MaS[4][LDJ * LDJ], MbS[4][LDJ * LDJ],
                     MvS[4][LDJ * LDJ], MwS[4][LDJ * LDJ];
    const int wv = threadIdx.x >> 5, lane = threadIdx.x & 31;
    const int idx = blockIdx.x * 4 + wv;
    const int b = idx / ND, d = idx - b * ND;
    float *Ma = MaS[wv], *Mb = MbS[wv], *Mv = MvS[wv], *Mw = MwS[wv];

    for (int i = lane; i < ND2 * ND2; i += 32) {
        Ma[(i / ND2) * LDJ + (i % ND2)] = S2g[(size_t)idx * (ND2 * ND2) + i];
        Mw[(i / ND2) * LDJ + (i % ND2)] = Gm[i];
    }
    WFENCE();
    // T = S2 * Gm -> Mb
    if (lane < ND2)
        for (int i = 0; i < ND2; ++i) {
            float s = 0.f;
            for (int t = 0; t < ND2; ++t) s += Ma[i * LDJ + t] * Mw[t * LDJ + lane];
            Mb[i * LDJ + lane] = s;
        }
    WFENCE();
    // Sf = Gm * T -> Ma
    if (lane < ND2)
        for (int i = 0; i < ND2; ++i) {
            float s = 0.f;
            for (int t = 0; t < ND2; ++t) s += Mw[i * LDJ + t] * Mb[t * LDJ + lane];
            Ma[i * LDJ + lane] = s;
        }
    WFENCE();
    jacobi_sym(Ma, Mv, ND2, lane);
    if (lane < ND2)
        for (int i = 0; i < ND2; ++i) {
            float s = 0.f;
            for (int t = 0; t < ND2; ++t)
                s += Mv[i * LDJ + t] * logf(fmaxf(Ma[t * LDJ + t], EPSV)) * Mv[lane * LDJ + t];
            Xlg[(size_t)b * (ND * ND2 * ND2) + d * (ND2 * ND2) + i * ND2 + lane] = s;
        }
}

// ---------------------------------------------------------------------------
// feat = Xv(256x11664) * conv_w^T(11664x16) + conv_b; out = feat * lin_w^T + lin_b
// 16 blocks x 4 waves; K=11664 split 4 ways (729 WMMA k-steps/wave), LDS reduce.
__global__ void feat_kernel(const float* __restrict__ Xl,
                            const float* __restrict__ conv_w,
                            const float* __restrict__ conv_b,
                            const float* __restrict__ lin_w,
                            const float* __restrict__ lin_b,
                            float* __restrict__ outp) {
    __shared__ float pf[4 * 256];
    const int lane = threadIdx.x & 31, wv = threadIdx.x >> 5;
    const int M = blockIdx.x * 16;
    const int m = lane & 15, ko = 2 * (lane >> 4);
    const int KW = ND * ND2 * ND2;    // 11664
    const int kk0 = wv * (KW / 16);   // 729 k-steps per wave
    const float* arow = Xl + (size_t)(M + m) * KW;
    const float* brow = conv_w + (size_t)m * KW;
    v8f acc = {};
    for (int kk = kk0; kk < kk0 + KW / 16; ++kk) {
        int k = kk * 4 + ko;
        if ((kk & 63) == 0) {
            __builtin_prefetch(arow + kk * 4 + 1024, 0, 1);
            __builtin_prefetch(brow + kk * 4 + 1024, 0, 1);
        }
        v2f a = *(const v2f*)&arow[k];
        v2f bb = *(const v2f*)&brow[k];
        acc = __builtin_amdgcn_wmma_f32_16x16x4_f32(false, a, false, bb, (short)0, acc, false, false);
    }
    const int nn = lane & 15, mhi = 8 * (lane >> 4);
    #pragma unroll
    for (int r = 0; r < 8; ++r) pf[wv * 256 + (mhi + r) * 16 + nn] = acc[r];
    __syncthreads();
    // reduce 4 partial tiles
    for (int e = threadIdx.x; e < 256; e += 128)
        pf[e] = pf[e] + pf[256 + e] + pf[512 + e] + pf[768 + e];
    __syncthreads();
    if (threadIdx.x < 64) {
        int row = threadIdx.x >> 2, o = threadIdx.x & 3;
        float s = lin_b[o];
        for (int k2 = 0; k2 < NTC; ++k2)
            s += (pf[row * 16 + k2] + conv_b[k2]) * lin_w[o * NTC + k2];
        outp[(M + row) * NOUT + o] = s;
    }
}

// ---------------------------------------------------------------------------
extern "C" void kernel_launch(void* const* d_in, const int* in_sizes, int n_in,
                              void* d_out, int out_size, void* d_ws, size_t ws_size,
                              hipStream_t stream) {
    const float* x      = (const float*)d_in[0];
    const float* W1     = (const float*)d_in[1];
    const float* W2     = (const float*)d_in[2];
    const float* conv_w = (const float*)d_in[3];
    const float* conv_b = (const float*)d_in[4];
    const float* lin_w  = (const float*)d_in[5];
    const float* lin_b  = (const float*)d_in[6];
    float* out = (float*)d_out;
    float* ws  = (float*)d_ws;

    float* covb = ws + WS_COV;
    float* S2b  = ws + WS_S2;
    float* Xlb  = ws + WS_XL;
    float* Glb  = ws + WS_GL;
    float* Gmb  = ws + WS_GM;

    zero_gl_kernel<<<1, 352, 0, stream>>>(Glb);
    cov_kernel<<<NMAT / 2, 64, 0, stream>>>(x, covb);
    bimap_kernel<<<NMAT / 4, 128, 0, stream>>>(covb, W1, W2, S2b, Glb);
    gm_kernel<<<1, 32, 0, stream>>>(Glb, Gmb);
    bn_kernel<<<NMAT / 4, 128, 0, stream>>>(S2b, Gmb, Xlb);
    feat_kernel<<<NB / 16, 32 * 4, 0, stream>>>(Xlb, conv_w, conv_b, lin_w, lin_b, out);
}